// HodgkinHuxley_86011015070338
// MI455X (gfx1250) — compile-verified
//
#include <hip/hip_runtime.h>
#include <math.h>

// ---------------------------------------------------------------------------
// Hodgkin-Huxley particle-filter scan for MI455X (gfx1250, wave32).
//   Pass 1: serial time scan, 1 wave per block (32 particles), 16 blocks.
//           CDNA5 async global->LDS double-buffered prefetch of eps_V / I
//           tiles (ASYNCcnt), LDS-staged coalesced b128 writeback of zs/nat.
//   Pass 2: elementwise xs/nat post-pass, fully coalesced.
// ---------------------------------------------------------------------------

#define T_STEPS 8192
#define N_PART  512
#define DT      0.001f
#define LOG_2PI 1.8378770664093453f

#define TB        64                 // time-tile
#define NT        (T_STEPS / TB)     // 128 tiles
#define EPS_PITCH 68                 // 64 + 4 pad: 16B-aligned rows, low-conflict column reads
#define Z_PITCH   65                 // float4 row pitch for z staging

#if defined(__HIP_DEVICE_COMPILE__) && defined(__gfx1250__) &&                 \
    __has_builtin(__builtin_amdgcn_global_load_async_to_lds_b128) &&           \
    __has_builtin(__builtin_amdgcn_global_load_async_to_lds_b64)
#define HAVE_ASYNC_LDS 1
#else
#define HAVE_ASYNC_LDS 0
#endif

#define AS1 __attribute__((address_space(1)))
#define AS3 __attribute__((address_space(3)))

typedef int v4i __attribute__((ext_vector_type(4)));
typedef int v2i __attribute__((ext_vector_type(2)));

#if HAVE_ASYNC_LDS
template <int N>
__device__ __forceinline__ void wait_async() {
#if __has_builtin(__builtin_amdgcn_s_wait_asynccnt)
  __builtin_amdgcn_s_wait_asynccnt(N);
#else
  asm volatile("s_wait_asynccnt %0" ::"i"(N) : "memory");
#endif
  asm volatile("" ::: "memory");  // keep LDS reads below the wait
}

__device__ __forceinline__ void issue_tile_async(const float* __restrict__ epsV,
                                                 const float* __restrict__ I,
                                                 float (*sEps)[EPS_PITCH],
                                                 float* sIrow, int p0, int t0,
                                                 int lane) {
  // eps_V tile: 32 rows x 64 floats = 512 float4s, 16 async b128 per lane.
  // Each instruction: 32 lanes x 16B, coalesced along a row in global memory.
#pragma unroll
  for (int i = 0; i < 16; ++i) {
    const int e   = i * 32 + lane;
    const int row = e >> 4;       // 16 float4 per row
    const int c4  = e & 15;
    const float* g = epsV + (size_t)(p0 + row) * T_STEPS + t0 + c4 * 4;
    float*       l = &sEps[row][c4 * 4];
    __builtin_amdgcn_global_load_async_to_lds_b128((AS1 v4i*)(g), (AS3 v4i*)(l),
                                                   0, 0);
  }
  // I tile: 64 floats via one async b64 per lane.
  __builtin_amdgcn_global_load_async_to_lds_b64(
      (AS1 v2i*)(I + t0 + lane * 2), (AS3 v2i*)(sIrow + lane * 2), 0, 0);
}
#endif  // HAVE_ASYNC_LDS

__device__ __forceinline__ void load_tile_sync(const float* __restrict__ epsV,
                                               const float* __restrict__ I,
                                               float (*sEps)[EPS_PITCH],
                                               float* sIrow, int p0, int t0,
                                               int lane) {
#pragma unroll
  for (int i = 0; i < 16; ++i) {
    const int e   = i * 32 + lane;
    const int row = e >> 4;
    const int c4  = e & 15;
    const float4 v =
        *(const float4*)(epsV + (size_t)(p0 + row) * T_STEPS + t0 + c4 * 4);
    *(float4*)&sEps[row][c4 * 4] = v;
  }
  sIrow[lane * 2]     = I[t0 + lane * 2];
  sIrow[lane * 2 + 1] = I[t0 + lane * 2 + 1];
  __syncthreads();
}

__global__ __launch_bounds__(32) void hh_scan_kernel(
    const float* __restrict__ I, const float* __restrict__ epsV,
    const float* __restrict__ z0, const float* __restrict__ p_lCm,
    const float* __restrict__ p_lgl, const float* __restrict__ p_lnEl,
    const float* __restrict__ p_lgna, const float* __restrict__ p_lEna,
    const float* __restrict__ p_lgk, const float* __restrict__ p_lnEk,
    const float* __restrict__ p_lnVt, const float* __restrict__ p_lrs,
    const float* __restrict__ p_lns, float* __restrict__ zs,
    float* __restrict__ nlp, float* __restrict__ nat) {
  __shared__ float  sEps[2][32][EPS_PITCH];  // 17408 B
  __shared__ float  sI[2][TB];               //   512 B
  __shared__ float4 sZ[32][Z_PITCH];         // 33280 B
  __shared__ float  sV[32][EPS_PITCH];       //  8704 B   (total 59904 B)

  const int lane = threadIdx.x;
  const int p0   = blockIdx.x * 32;
  const int p    = p0 + lane;

  // Parameters (uniform; one-time scalar loads + exp).
  const float log_rs  = *p_lrs;
  const float log_ns  = *p_lns;
  const float C_m     = expf(*p_lCm);
  const float g_l     = expf(*p_lgl);
  const float E_l     = -expf(*p_lnEl);
  const float g_na    = expf(*p_lgna);
  const float E_na    = expf(*p_lEna);
  const float g_k     = expf(*p_lgk);
  const float E_k     = -expf(*p_lnEk);
  const float V_t     = -expf(*p_lnVt);
  const float rand_std  = expf(log_rs);
  const float noise_std = expf(log_ns);
  const float inv_var   = 1.0f / (noise_std * noise_std);
  const float dt_over_C = DT / C_m;
  const float lp_c      = -log_rs - 0.5f * LOG_2PI;  // logp per-step constant
  const float nm_c      = log_ns + 0.5f * LOG_2PI;   // norm per-step constant
  const float nat1      = -0.5f * inv_var;

  float V = z0[0], m = z0[1], h = z0[2], n = z0[3];
  float logp = 0.0f, norm = 0.0f;

#if HAVE_ASYNC_LDS
  issue_tile_async(epsV, I, sEps[0], sI[0], p0, 0, lane);
  issue_tile_async(epsV, I, sEps[1], sI[1], p0, TB, lane);
#endif

  for (int k = 0; k < NT; ++k) {
    const int buf = k & 1;
    const int t0  = k * TB;

#if HAVE_ASYNC_LDS
    // 17 loads per tile; in-order completion => cnt<=17 means tile k landed.
    if (k + 1 < NT) wait_async<17>();
    else            wait_async<0>();
#else
    load_tile_sync(epsV, I, sEps[buf], sI[buf], p0, t0, lane);
#endif

    const float* eRow = &sEps[buf][lane][0];
    const float* iRow = &sI[buf][0];
    for (int t = 0; t < TB; ++t) {
      const float eps = eRow[t];
      const float I_t = iRow[t];

      const float u  = V - V_t;
      const float x1 = u - 13.0f;
      const float x2 = u - 40.0f;
      const float x3 = u - 15.0f;
      // 5 v_exp_f32 + 4 fast reciprocals (TRANS co-executes with VALU).
      const float e1 = __expf(x1 * -0.25f);          // exp(-(u-13)/4)
      const float e2 = __expf(x2 * -0.2f);           // exp(-(u-40)/5)
      const float e3 = __expf(x3 * -0.2f);           // exp(-(u-15)/5)
      const float a_m = __fdividef(-0.32f * x1, e1 - 1.0f);
      const float b_m = __fdividef(0.28f * x2 * e2, 1.0f - e2);  // 0.28*x2/(1/e2-1)
      const float a_h = 0.128f * __expf((u - 17.0f) * (-1.0f / 18.0f));
      const float b_h = __fdividef(4.0f, e2 + 1.0f);
      const float a_n = __fdividef(-0.032f * x3, e3 - 1.0f);
      const float b_n = 0.5f * __expf((u - 10.0f) * (-1.0f / 40.0f));

      const float dm = DT * (a_m * (1.0f - m) - b_m * m);
      const float dh = DT * (a_h * (1.0f - h) - b_h * h);
      const float dn = DT * (a_n * (1.0f - n) - b_n * n);

      const float I_na = g_na * m * m * m * h * (V - E_na);
      const float n2   = n * n;
      const float I_k  = g_k * n2 * n2 * (V - E_k);
      const float dV   = dt_over_C * (-g_l * (V - E_l) - I_na - I_k + I_t);

      m += dm; h += dh; n += dn;
      const float Vn = V + dV + rand_std * eps;

      logp += fmaf(-0.5f * eps, eps, lp_c);
      norm += fmaf(0.5f * inv_var * Vn, Vn, nm_c);

      sZ[lane][t] = make_float4(Vn, m, h, n);
      sV[lane][t] = Vn;
      V = Vn;
    }
    __syncthreads();  // single-wave WG: barrier is ~free, orders LDS

    // Coalesced writeback: zs tile as b128 (lanes cover consecutive t).
#pragma unroll 8
    for (int i = 0; i < 64; ++i) {
      const int e = i * 32 + lane;
      const int r = e >> 6;
      const int t = e & 63;
      *(float4*)(zs + (((size_t)(p0 + r) * T_STEPS + t0 + t) << 2)) = sZ[r][t];
    }
    // nat tile as b64: (Vs*inv_var, -0.5*inv_var).
#pragma unroll 8
    for (int i = 0; i < 64; ++i) {
      const int e = i * 32 + lane;
      const int r = e >> 6;
      const int t = e & 63;
      const float2 v = make_float2(sV[r][t] * inv_var, nat1);
      *(float2*)(nat + (((size_t)(p0 + r) * T_STEPS + t0 + t) << 1)) = v;
    }
    __syncthreads();

#if HAVE_ASYNC_LDS
    if (k + 2 < NT)
      issue_tile_async(epsV, I, sEps[buf], sI[buf], p0, t0 + 2 * TB, lane);
#endif
  }

  nlp[p] = norm - logp;
}

// Pass 2: xs = Vs + noise_std*eps_x, recovering Vs = nat0 * noise_std^2.
__global__ __launch_bounds__(256) void hh_post_kernel(
    const float* __restrict__ epsX, const float* __restrict__ nat,
    const float* __restrict__ p_lns, float* __restrict__ xs) {
  const float noise_std = expf(*p_lns);
  const float var       = noise_std * noise_std;
  const size_t j = (size_t)blockIdx.x * blockDim.x + threadIdx.x;  // 2 elems
  if (j < (size_t)N_PART * T_STEPS / 2) {
    const float4 nv = *(const float4*)(nat + j * 4);   // nat[2j], nat[2j+1]
    const float2 ev = *(const float2*)(epsX + j * 2);
    const float2 o  = make_float2(fmaf(noise_std, ev.x, nv.x * var),
                                  fmaf(noise_std, ev.y, nv.z * var));
    *(float2*)(xs + j * 2) = o;
  }
}

extern "C" void kernel_launch(void* const* d_in, const int* in_sizes, int n_in,
                              void* d_out, int out_size, void* d_ws,
                              size_t ws_size, hipStream_t stream) {
  (void)in_sizes; (void)n_in; (void)out_size; (void)d_ws; (void)ws_size;
  const float* I    = (const float*)d_in[0];
  const float* epsV = (const float*)d_in[1];
  const float* epsX = (const float*)d_in[2];
  const float* z0   = (const float*)d_in[3];
  const float* lCm  = (const float*)d_in[4];
  const float* lgl  = (const float*)d_in[5];
  const float* lnEl = (const float*)d_in[6];
  const float* lgna = (const float*)d_in[7];
  const float* lEna = (const float*)d_in[8];
  const float* lgk  = (const float*)d_in[9];
  const float* lnEk = (const float*)d_in[10];
  const float* lnVt = (const float*)d_in[11];
  const float* lrs  = (const float*)d_in[12];
  const float* lns  = (const float*)d_in[13];

  float* out = (float*)d_out;
  float* zs  = out;                                    // (N, T, 4)
  float* xs  = zs + (size_t)N_PART * T_STEPS * 4;      // (N, T)
  float* nlp = xs + (size_t)N_PART * T_STEPS;          // (N,)
  float* nat = nlp + N_PART;                           // (N, T, 2)

  hh_scan_kernel<<<N_PART / 32, 32, 0, stream>>>(
      I, epsV, z0, lCm, lgl, lnEl, lgna, lEna, lgk, lnEk, lnVt, lrs, lns, zs,
      nlp, nat);

  const int ne = N_PART * T_STEPS / 2;
  hh_post_kernel<<<(ne + 255) / 256, 256, 0, stream>>>(epsX, nat, lns, xs);
}